// QuantumNATHybrid_65481071400279
// MI455X (gfx1250) — compile-verified
//
#include <hip/hip_runtime.h>
#include <math.h>

#define N_WIRES 4
#define QDIM    16
#define BATCH   32768
#define BLOCK   256
#define NBLK    (BATCH / BLOCK)   // 128
#define EPSV    1e-5f

typedef float v2f __attribute__((ext_vector_type(2)));
typedef float v8f __attribute__((ext_vector_type(8)));

// ---------------------------------------------------------------------------
// Kernel 0: build fused per-(layer,wire) 2x2 complex unitaries U = Uz*Uy*Ux
// 8 gates x 8 floats = 64 floats -> ws. All trig happens here, once.
// ---------------------------------------------------------------------------
struct Cplx { float r, i; };
__device__ __forceinline__ Cplx cmul(Cplx a, Cplx b) {
    return { a.r * b.r - a.i * b.i, a.r * b.i + a.i * b.r };
}
__device__ __forceinline__ Cplx cadd(Cplx a, Cplx b) { return { a.r + b.r, a.i + b.i }; }

__global__ __launch_bounds__(8) void k_gates(const float* __restrict__ qw,
                                             float* __restrict__ gates)
{
    const int g = threadIdx.x;            // 0..7 = (layer*4 + wire)
    const float tx = qw[g * 3 + 0], ty = qw[g * 3 + 1], tz = qw[g * 3 + 2];
    const float cx = cosf(0.5f * tx), sx = sinf(0.5f * tx);
    const float cy = cosf(0.5f * ty), sy = sinf(0.5f * ty);
    const float cz = cosf(0.5f * tz), sz = sinf(0.5f * tz);

    // Ux = [[cx, -i sx], [-i sx, cx]];  Uy = [[cy, -sy], [sy, cy]]
    Cplx x00{cx, 0.f}, x01{0.f, -sx}, x10{0.f, -sx}, x11{cx, 0.f};
    Cplx y00{cy, 0.f}, y01{-sy, 0.f}, y10{sy, 0.f}, y11{cy, 0.f};
    // U1 = Uy * Ux
    Cplx u00 = cadd(cmul(y00, x00), cmul(y01, x10));
    Cplx u01 = cadd(cmul(y00, x01), cmul(y01, x11));
    Cplx u10 = cadd(cmul(y10, x00), cmul(y11, x10));
    Cplx u11 = cadd(cmul(y10, x01), cmul(y11, x11));
    // U = diag(cz - i sz, cz + i sz) * U1
    Cplx z0{cz, -sz}, z1{cz, sz};
    u00 = cmul(z0, u00);  u01 = cmul(z0, u01);
    u10 = cmul(z1, u10);  u11 = cmul(z1, u11);

    float* o = gates + g * 8;
    o[0] = u00.r; o[1] = u00.i; o[2] = u01.r; o[3] = u01.i;
    o[4] = u10.r; o[5] = u10.i; o[6] = u11.r; o[7] = u11.i;
}

// ---------------------------------------------------------------------------
// Kernel 1: pooling -> quantum circuit -> WMMA MLP -> pre-norm out + partials
// ---------------------------------------------------------------------------
__global__ __launch_bounds__(BLOCK) void k_main(
    const float* __restrict__ x,  const float* __restrict__ gates,
    const float* __restrict__ w1, const float* __restrict__ b1,
    const float* __restrict__ w2, const float* __restrict__ b2,
    float* __restrict__ out, float* __restrict__ partials)
{
    __shared__ float lds_q[BLOCK][4];     // qout per sample
    __shared__ float lds_h[8][16 * 64];   // per-wave hidden tile (16x64)
    __shared__ float s_sum[BLOCK][4];
    __shared__ float s_sq[BLOCK][4];
    __shared__ float lw1[4 * 64];
    __shared__ float lw2[64 * 4];
    __shared__ float lb1[64];
    __shared__ float lb2[4];
    __shared__ float lg[64];              // fused gate matrices

    const int tid = threadIdx.x;

    // stage weights + gates into LDS
    lw1[tid] = w1[tid];                    // 256 elements
    lw2[tid] = w2[tid];                    // 256 elements
    if (tid < 64) { lb1[tid] = b1[tid]; lg[tid] = gates[tid]; }
    if (tid < 4)  lb2[tid] = b2[tid];
    __syncthreads();

    // ---------------- pooling: first row of 4x4 pool grid ----------------
    const int b = blockIdx.x * BLOCK + tid;
    const float* xs = x + (size_t)b * 784;   // 28*28
    float f[4] = {0.f, 0.f, 0.f, 0.f};
    #pragma unroll
    for (int r = 0; r < 6; ++r) {
        #pragma unroll
        for (int j = 0; j < 4; ++j) {
            #pragma unroll
            for (int c = 0; c < 6; ++c)
                f[j] += xs[r * 28 + j * 6 + c];
        }
    }
    #pragma unroll
    for (int j = 0; j < 4; ++j) f[j] *= (1.0f / 36.0f);

    // ---------------- quantum circuit (16 complex amplitudes) -------------
    float nrm = sqrtf(f[0]*f[0] + f[1]*f[1] + f[2]*f[2] + f[3]*f[3]);
    float re[QDIM], im[QDIM];
    #pragma unroll
    for (int i = 0; i < QDIM; ++i) { re[i] = (i < 4) ? f[i] / nrm : 0.0f; im[i] = 0.0f; }

    #pragma unroll
    for (int l = 0; l < 2; ++l) {
        #pragma unroll
        for (int w = 0; w < 4; ++w) {
            const float* g = &lg[(l * 4 + w) * 8];
            const float u00r = g[0], u00i = g[1], u01r = g[2], u01i = g[3];
            const float u10r = g[4], u10i = g[5], u11r = g[6], u11i = g[7];
            const int R = 1 << (3 - w);      // wire w = bit (3-w)
            #pragma unroll
            for (int i0 = 0; i0 < QDIM; ++i0) {
                if (i0 & R) continue;
                const int i1 = i0 | R;
                const float r0 = re[i0], m0 = im[i0], r1 = re[i1], m1 = im[i1];
                re[i0] = u00r * r0 - u00i * m0 + u01r * r1 - u01i * m1;
                im[i0] = u00r * m0 + u00i * r0 + u01r * m1 + u01i * r1;
                re[i1] = u10r * r0 - u10i * m0 + u11r * r1 - u11i * m1;
                im[i1] = u10r * m0 + u10i * r0 + u11r * m1 + u11i * r1;
            }
        }
        // CZ ring: pairs (0,1)(1,2)(2,3)(3,0); combined parity sign
        #pragma unroll
        for (int i = 0; i < QDIM; ++i) {
            const int b0 = (i >> 3) & 1, b1_ = (i >> 2) & 1, b2_ = (i >> 1) & 1, b3 = i & 1;
            if ((b0 & b1_) ^ (b1_ & b2_) ^ (b2_ & b3) ^ (b3 & b0)) { re[i] = -re[i]; im[i] = -im[i]; }
        }
    }
    // Z expectations
    float q[4] = {0.f, 0.f, 0.f, 0.f};
    #pragma unroll
    for (int i = 0; i < QDIM; ++i) {
        const float p = re[i] * re[i] + im[i] * im[i];
        #pragma unroll
        for (int k = 0; k < 4; ++k)
            q[k] += p * (1.0f - 2.0f * (float)((i >> (3 - k)) & 1));
    }
    #pragma unroll
    for (int k = 0; k < 4; ++k) lds_q[tid][k] = q[k];
    __syncthreads();

    // ---------------- WMMA MLP: h = relu(q@w1+b1); out = h@w2+b2 ----------
    const int wid  = tid >> 5;       // wave 0..7
    const int lane = tid & 31;
    const int hh   = lane >> 4;      // lane half (K-pair select)
    const int nn   = lane & 15;      // N (and M for A-load)
    float* hbuf = &lds_h[wid][0];

    for (int t2 = 0; t2 < 2; ++t2) {
        const int tile = wid * 2 + t2;          // 16-sample tile
        // A: 16x4 of qout.  lanes<16: K=0,1 ; lanes>=16: K=2,3
        v2f a;
        a.x = lds_q[tile * 16 + nn][2 * hh];
        a.y = lds_q[tile * 16 + nn][2 * hh + 1];

        // GEMM1: four N-tiles of w1 (4x64)
        #pragma unroll
        for (int t = 0; t < 4; ++t) {
            const int col = t * 16 + nn;
            v2f bv;
            bv.x = lw1[(2 * hh) * 64 + col];
            bv.y = lw1[(2 * hh + 1) * 64 + col];
            v8f c = {};
            c = __builtin_amdgcn_wmma_f32_16x16x4_f32(false, a, false, bv,
                                                      (short)0, c, false, false);
            const float bias = lb1[col];
            #pragma unroll
            for (int v = 0; v < 8; ++v) {
                float hv = c[v] + bias;
                hv = hv > 0.0f ? hv : 0.0f;
                hbuf[(v + 8 * hh) * 64 + col] = hv;   // C-layout -> row-major LDS
            }
        }
        __syncthreads();

        // GEMM2: out(16x4) = h(16x64) @ w2(64x4), K in steps of 4, N padded to 16
        v8f acc = {};
        #pragma unroll
        for (int k = 0; k < 16; ++k) {
            v2f a2;
            a2.x = hbuf[nn * 64 + 4 * k + 2 * hh];
            a2.y = hbuf[nn * 64 + 4 * k + 2 * hh + 1];
            v2f b2v;
            b2v.x = (nn < 4) ? lw2[(4 * k + 2 * hh) * 4 + nn] : 0.0f;
            b2v.y = (nn < 4) ? lw2[(4 * k + 2 * hh + 1) * 4 + nn] : 0.0f;
            acc = __builtin_amdgcn_wmma_f32_16x16x4_f32(false, a2, false, b2v,
                                                        (short)0, acc, false, false);
        }
        // scatter valid columns (n<4): lane holds out[v+8*hh][nn]
        #pragma unroll
        for (int v = 0; v < 8; ++v) {
            if (nn < 4) {
                const int s = tile * 16 + v + 8 * hh;
                const float val = acc[v] + lb2[nn];
                out[(size_t)(blockIdx.x * BLOCK + s) * 4 + nn] = val;
                s_sum[s][nn] = val;
                s_sq[s][nn]  = val * val;
            }
        }
        __syncthreads();
    }

    // deterministic block tree-reduction of sums / sumsq
    for (int off = 128; off > 0; off >>= 1) {
        if (tid < off) {
            #pragma unroll
            for (int c = 0; c < 4; ++c) {
                s_sum[tid][c] += s_sum[tid + off][c];
                s_sq[tid][c]  += s_sq[tid + off][c];
            }
        }
        __syncthreads();
    }
    if (tid < 8)
        partials[blockIdx.x * 8 + tid] = (tid < 4) ? s_sum[0][tid] : s_sq[0][tid - 4];
}

// ---------------------------------------------------------------------------
// Kernel 2: reduce per-block partials -> mu, inv_std
// ---------------------------------------------------------------------------
__global__ __launch_bounds__(128) void k_stats(const float* __restrict__ partials,
                                               float* __restrict__ stats)
{
    __shared__ float sp[128][8];
    const int tid = threadIdx.x;
    #pragma unroll
    for (int c = 0; c < 8; ++c) sp[tid][c] = partials[tid * 8 + c];
    __syncthreads();
    for (int off = 64; off > 0; off >>= 1) {
        if (tid < off) {
            #pragma unroll
            for (int c = 0; c < 8; ++c) sp[tid][c] += sp[tid + off][c];
        }
        __syncthreads();
    }
    if (tid < 4) {
        const float mu  = sp[0][tid] / (float)BATCH;
        const float var = sp[0][tid + 4] / (float)BATCH - mu * mu;
        stats[tid]     = mu;
        stats[4 + tid] = rsqrtf(var + EPSV);
    }
}

// ---------------------------------------------------------------------------
// Kernel 3: in-place batchnorm affine
// ---------------------------------------------------------------------------
__global__ __launch_bounds__(256) void k_norm(float* __restrict__ out,
                                              const float* __restrict__ stats,
                                              const float* __restrict__ gamma,
                                              const float* __restrict__ beta)
{
    const int i = blockIdx.x * 256 + threadIdx.x;
    const int c = i & 3;
    const float v = out[i];
    out[i] = (v - stats[c]) * stats[4 + c] * gamma[c] + beta[c];
}

// ---------------------------------------------------------------------------
extern "C" void kernel_launch(void* const* d_in, const int* in_sizes, int n_in,
                              void* d_out, int out_size, void* d_ws, size_t ws_size,
                              hipStream_t stream)
{
    const float* x     = (const float*)d_in[0];
    const float* qw    = (const float*)d_in[1];
    const float* w1    = (const float*)d_in[2];
    const float* b1    = (const float*)d_in[3];
    const float* w2    = (const float*)d_in[4];
    const float* b2    = (const float*)d_in[5];
    const float* gamma = (const float*)d_in[6];
    const float* beta  = (const float*)d_in[7];
    float* out      = (float*)d_out;
    float* gates    = (float*)d_ws;            // 64 floats
    float* partials = gates + 64;              // NBLK*8 floats
    float* stats    = partials + NBLK * 8;     // 8 floats

    k_gates<<<1, 8, 0, stream>>>(qw, gates);
    k_main <<<NBLK, BLOCK, 0, stream>>>(x, gates, w1, b1, w2, b2, out, partials);
    k_stats<<<1, 128, 0, stream>>>(partials, stats);
    k_norm <<<(BATCH * 4) / 256, 256, 0, stream>>>(out, stats, gamma, beta);
}